// projector_43138651521628
// MI455X (gfx1250) — compile-verified
//
#include <hip/hip_runtime.h>
#include <hip/hip_bf16.h>

#define HH 512
#define WW 512
#define NPIX (HH * WW)
#define BB 4
#define CCH 16

typedef __attribute__((ext_vector_type(2))) float v2f;
typedef __attribute__((ext_vector_type(8))) float v8f;

// ---------------------------------------------------------------------------
// Kernel 1: out = depth (float4 copy). Kernel boundary orders this before the
// scatter writes of kernel 2.
// ---------------------------------------------------------------------------
__global__ void projector_copy_kernel(const float4* __restrict__ in,
                                      float4* __restrict__ out, int n4) {
    int i = blockIdx.x * blockDim.x + threadIdx.x;
    if (i < n4) out[i] = in[i];
}

// ---------------------------------------------------------------------------
// Kernel 2: unproject -> per-channel 4x4 transforms via V_WMMA_F32_16X16X4_F32
// -> mask-select (last channel wins) -> reproject -> scatter Z.
//
// Wave layout: each wave32 owns 32 pixels = two 16-pixel M-tiles.
//   A (16x4 f32, 2 VGPRs): lanes 0-15 hold (K=0,K=1)=(x,y) of pixel M=lane,
//                          lanes 16-31 hold (K=2,K=3)=(z,1) of pixel M=lane-16.
//   B (4x16 f32, 2 VGPRs): column n = 4*chInGroup + comp; lanes 0-15 hold
//                          rows K=0/1, lanes 16-31 rows K=2/3. Four channel
//                          groups (g=0..3) cover all 16 channels.
//   D (16x16 f32, 8 VGPRs): VGPR j = row M=j (lanes 0-15) / M=j+8 (16-31).
// D tiles are parked in wave-private LDS, then phase 2 (one pixel per lane)
// does the masked select + divide + reproject + scatter.
// ---------------------------------------------------------------------------
__global__ __launch_bounds__(128) void projector_43138651521628_kernel(
    const float* __restrict__ depth,   // [B,1,H,W]
    const float* __restrict__ Kmat,    // [B,3,3]
    const float* __restrict__ Tmat,    // [B,C,4,4]
    const int*   __restrict__ masks,   // [B,C,H,W]
    float* __restrict__ out)           // [B,1,H,W], pre-initialized = depth
{
    __shared__ float lds[4][32][CCH * 4];   // [wave][pixelLocal][ch*4+comp] = 32 KB

    const int b    = blockIdx.y;
    const int tid  = threadIdx.x;
    const int wave = tid >> 5;
    const int lane = tid & 31;
    const int half = lane >> 4;       // 0: lanes 0-15, 1: lanes 16-31
    const int l16  = lane & 15;
    const int pixBase = blockIdx.x * 128 + wave * 32;   // within batch b

    // ---- K and analytic K^-1 (cofactors; uniform per block -> scalarized) ----
    const float* Kb = Kmat + b * 9;
    const float k00 = Kb[0], k01 = Kb[1], k02 = Kb[2];
    const float k10 = Kb[3], k11 = Kb[4], k12 = Kb[5];
    const float k20 = Kb[6], k21 = Kb[7], k22 = Kb[8];
    const float det = k00 * (k11 * k22 - k12 * k21)
                    - k01 * (k10 * k22 - k12 * k20)
                    + k02 * (k10 * k21 - k11 * k20);
    const float rd = 1.0f / det;
    const float i00 =  (k11 * k22 - k12 * k21) * rd;
    const float i01 = -(k01 * k22 - k02 * k21) * rd;
    const float i02 =  (k01 * k12 - k02 * k11) * rd;
    const float i10 = -(k10 * k22 - k12 * k20) * rd;
    const float i11 =  (k00 * k22 - k02 * k20) * rd;
    const float i12 = -(k00 * k12 - k02 * k10) * rd;
    const float i20 =  (k10 * k21 - k11 * k20) * rd;
    const float i21 = -(k00 * k21 - k01 * k20) * rd;
    const float i22 =  (k00 * k11 - k01 * k10) * rd;

    // ---- Phase 1a: unproject; build A operands for the two 16-pixel tiles ----
    float ptx[2], pty[2], ptz[2];
    v2f A[2];
    #pragma unroll
    for (int t = 0; t < 2; ++t) {
        const int p = pixBase + t * 16 + l16;           // 0..NPIX-1
        const float fu = (float)(p & (WW - 1));
        const float fv = (float)(p >> 9);
        const float d  = depth[b * NPIX + p];
        const float x = (i00 * fu + i01 * fv + i02) * d;
        const float y = (i10 * fu + i11 * fv + i12) * d;
        const float z = (i20 * fu + i21 * fv + i22) * d;
        ptx[t] = x; pty[t] = y; ptz[t] = z;
        A[t].x = half ? z    : x;    // K = 2*half + 0
        A[t].y = half ? 1.0f : y;    // K = 2*half + 1
    }

    // ---- Phase 1b: 4 channel-groups x 2 tiles of WMMA, results -> LDS ----
    const float* Tb = Tmat + b * CCH * 16;
    #pragma unroll
    for (int g = 0; g < 4; ++g) {
        const int ch   = 4 * g + (l16 >> 2);   // channel for this B column
        const int comp = l16 & 3;              // output component (row of T)
        const float* Trow = Tb + ch * 16 + comp * 4;
        v2f Bm;
        Bm.x = Trow[2 * half + 0];             // K = 2*half + 0
        Bm.y = Trow[2 * half + 1];             // K = 2*half + 1
        #pragma unroll
        for (int t = 0; t < 2; ++t) {
            v8f c = {};
            v8f D = __builtin_amdgcn_wmma_f32_16x16x4_f32(
                false, A[t], false, Bm, (short)0, c, false, false);
            #pragma unroll
            for (int j = 0; j < 8; ++j) {
                const int m = t * 16 + (half ? j + 8 : j);   // local pixel row
                lds[wave][m][g * 16 + l16] = D[j];
            }
        }
    }

    // Same wave produces & consumes its LDS region: wave-local DS ordering.
    asm volatile("s_wait_dscnt 0" ::: "memory");

    // ---- Phase 2: one pixel per lane: select, divide, reproject, scatter ----
    const int p = pixBase + lane;
    const int* mb = masks + (b * CCH) * NPIX + p;
    int sel = -1;
    #pragma unroll
    for (int ci = 0; ci < CCH; ++ci) {
        if (mb[ci * NPIX] != 0) sel = ci;      // later channel wins
    }

    // This lane already computed its own pixel's original point:
    // lanes 0-15 -> tile 0, lanes 16-31 -> tile 1.
    float tx = half ? ptx[1] : ptx[0];
    float ty = half ? pty[1] : pty[0];
    float tz = half ? ptz[1] : ptz[0];
    if (sel >= 0) {
        const float xh = lds[wave][lane][sel * 4 + 0];
        const float yh = lds[wave][lane][sel * 4 + 1];
        const float zh = lds[wave][lane][sel * 4 + 2];
        const float wh = lds[wave][lane][sel * 4 + 3];
        const float rw = 1.0f / (wh + 1e-10f);
        tx = xh * rw; ty = yh * rw; tz = zh * rw;
    }

    // Reproject with K, clip, scatter Z (plain store; duplicate-index races
    // resolve arbitrarily, matching the reference scatter's unspecified order).
    const float px = k00 * tx + k01 * ty + k02 * tz;
    const float py = k10 * tx + k11 * ty + k12 * tz;
    const float pz = k20 * tx + k21 * ty + k22 * tz;
    const float inv = 1.0f / (pz + 1e-10f);
    const int ui = (int)fminf(fmaxf(px * inv, 0.0f), (float)(WW - 1));
    const int vi = (int)fminf(fmaxf(py * inv, 0.0f), (float)(HH - 1));
    out[b * NPIX + vi * WW + ui] = tz;
}

extern "C" void kernel_launch(void* const* d_in, const int* in_sizes, int n_in,
                              void* d_out, int out_size, void* d_ws, size_t ws_size,
                              hipStream_t stream) {
    const float* depth = (const float*)d_in[0];   // [B,1,H,W] f32
    const float* Kmat  = (const float*)d_in[1];   // [B,3,3]   f32
    const float* Tmat  = (const float*)d_in[2];   // [B,C,4,4] f32
    const int*   masks = (const int*)d_in[3];     // [B,C,H,W] bool->int
    float* out = (float*)d_out;                   // [B,1,H,W] f32

    // 1) out = depth
    const int n4 = (BB * NPIX) / 4;
    projector_copy_kernel<<<(n4 + 255) / 256, 256, 0, stream>>>(
        (const float4*)depth, (float4*)out, n4);

    // 2) transform + scatter: 128 threads (4 waves x 32 pixels) per block
    dim3 grid(NPIX / 128, BB);
    projector_43138651521628_kernel<<<grid, 128, 0, stream>>>(
        depth, Kmat, Tmat, masks, out);
}